// HoughVotingModule_37194416783729
// MI455X (gfx1250) — compile-verified
//
#include <hip/hip_runtime.h>
#include <math.h>

// ---------------- problem constants (from reference) ----------------
#define RES_F      0.03f
#define NUM_ROTS   36
#define NUM_PROP   256
#define OVERSAMPLE 4
#define KSAMP      (NUM_PROP * OVERSAMPLE)   // 1024
#define GX 272
#define GY 112
#define GZ 272
#define NXZ (GX * GZ)                        // 73984
#define SEED_THRESH 0.3f
#define TWO_PI 6.28318530717958647692f

typedef __attribute__((ext_vector_type(2))) float v2f;
typedef __attribute__((ext_vector_type(8))) float v8f;

// ---------------- deterministic hash -> uniform(0,1) ----------------
__device__ __forceinline__ float hash_u01(unsigned long long x) {
    x ^= x >> 33; x *= 0xff51afd7ed558ccdULL;
    x ^= x >> 33; x *= 0xc4ceb9fe1a85ec53ULL;
    x ^= x >> 33;
    unsigned u = (unsigned)(x >> 40);                 // 24 random bits
    return ((float)u + 0.5f) * (1.0f / 16777216.0f);  // (0,1)
}

// ---------------- 1) voting: 1 thread per point, loop 36 rotations ----------------
__global__ void vote_kernel(const float* __restrict__ pc,
                            const float* __restrict__ xyz,
                            const float* __restrict__ scale,
                            const float* __restrict__ prob,
                            const float* __restrict__ corners,
                            float* __restrict__ grid_obj,
                            float* __restrict__ gsx,
                            float* __restrict__ gsy,
                            float* __restrict__ gsz,
                            int N) {
    int i = blockIdx.x * blockDim.x + threadIdx.x;
    if (i >= N) return;

    // streaming prefetch of upcoming point data (global_prefetch_b8)
    __builtin_prefetch(pc  + (size_t)(i + 4096) * 3, 0, 1);
    __builtin_prefetch(xyz + (size_t)(i + 4096) * 3, 0, 1);

    float px = pc[i * 3 + 0], py = pc[i * 3 + 1], pz = pc[i * 3 + 2];
    float ox = xyz[i * 3 + 0], oy = xyz[i * 3 + 1], oz = xyz[i * 3 + 2];
    float p  = prob[i];
    float c0x = corners[0], c0y = corners[1], c0z = corners[2];

    // y is rotation-invariant: hoist its bound check out of the loop
    float ty = py + oy;
    int iy = (int)floorf((ty - c0y) / RES_F + 0.5f);
    if (iy < 0 || iy >= GY) return;

    float wsx = p * scale[i * 3 + 0];
    float wsy = p * scale[i * 3 + 1];
    float wsz = p * scale[i * 3 + 2];

    #pragma unroll 4
    for (int r = 0; r < NUM_ROTS; ++r) {
        float th = TWO_PI * (float)r * (1.0f / (float)NUM_ROTS);
        float s, c;
        __sincosf(th, &s, &c);
        float tx = px + ox * c + oz * s;
        float tz = pz - ox * s + oz * c;
        int ix = (int)floorf((tx - c0x) / RES_F + 0.5f);
        int iz = (int)floorf((tz - c0z) / RES_F + 0.5f);
        if (ix < 0 || ix >= GX || iz < 0 || iz >= GZ) continue;  // w==0 -> skip
        size_t idx = ((size_t)ix * GY + iy) * GZ + iz;
        atomicAdd(&grid_obj[idx], p);     // L2-resident atomics (133MB < 192MB L2)
        atomicAdd(&gsx[idx], wsx);
        atomicAdd(&gsy[idx], wsy);
        atomicAdd(&gsz[idx], wsz);
    }
}

// ---------------- 2) max / argmax over y, then sqrt(max + 1e-7) ----------------
__global__ void reduce_y_kernel(const float* __restrict__ grid_obj,
                                float* __restrict__ dist,
                                int* __restrict__ yidx) {
    int t = blockIdx.x * blockDim.x + threadIdx.x;   // t = x*GZ + z
    if (t >= NXZ) return;
    int x = t / GZ, z = t % GZ;
    const float* base = grid_obj + (size_t)x * GY * GZ + z;
    float best = base[0];
    int bi = 0;
    #pragma unroll 4
    for (int y = 1; y < GY; ++y) {
        float v = base[(size_t)y * GZ];
        if (v > best) { best = v; bi = y; }          // first max (jnp.argmax)
    }
    dist[t] = sqrtf(best + 1e-7f);                   // POW = 0.5
    yidx[t] = bi;
}

// ---------------- 3) Gumbel-max categorical sample; one block per sample ----------------
__global__ void sample_kernel(const float* __restrict__ dist,
                              const int* __restrict__ yidx,
                              const float* __restrict__ corners,
                              const float* __restrict__ grid_obj,
                              const float* __restrict__ gsx,
                              const float* __restrict__ gsy,
                              const float* __restrict__ gsz,
                              float* __restrict__ wl4,   // [KSAMP][4] padded for WMMA
                              float* __restrict__ sc) {  // [KSAMP][3]
    int k = blockIdx.x;
    int t = threadIdx.x;

    float best = -3.4e38f;
    int bj = 0;
    for (int j = t; j < NXZ; j += blockDim.x) {
        unsigned long long key =
            ((unsigned long long)(unsigned)k << 32) ^ (unsigned long long)(unsigned)j
            ^ 0x9E3779B97F4A7C15ULL;
        float u = hash_u01(key);
        float g = -logf(-logf(u));                    // Gumbel(0,1)
        float s = logf(dist[j] + 1e-30f) + g;
        if (s > best) { best = s; bj = j; }
    }

    __shared__ float sv[256];
    __shared__ int   si[256];
    sv[t] = best; si[t] = bj;
    __syncthreads();
    for (int o = 128; o > 0; o >>= 1) {
        if (t < o) {
            if (sv[t + o] > sv[t] || (sv[t + o] == sv[t] && si[t + o] < si[t])) {
                sv[t] = sv[t + o]; si[t] = si[t + o];
            }
        }
        __syncthreads();
    }

    if (t == 0) {
        int s  = si[0];
        int xi = s / GZ, zi = s % GZ;
        int yi = yidx[s];
        float c0x = corners[0], c0y = corners[1], c0z = corners[2];
        wl4[k * 4 + 0] = (float)xi * RES_F + c0x;
        wl4[k * 4 + 1] = (float)yi * RES_F + c0y;
        wl4[k * 4 + 2] = (float)zi * RES_F + c0z;
        wl4[k * 4 + 3] = 0.0f;                       // K=4 pad for WMMA
        size_t idx = ((size_t)xi * GY + yi) * GZ + zi;
        float den = grid_obj[idx] + 1e-7f;
        sc[k * 3 + 0] = gsx[idx] / den;
        sc[k * 3 + 1] = gsy[idx] / den;
        sc[k * 3 + 2] = gsz[idx] / den;
    }
}

// ---------------- 4) min pairwise distance via V_WMMA_F32_16X16X4_F32 ----------------
// ||a-b||^2 = ||a||^2 + ||b||^2 - 2 a.b ; a.b done as 16x16x4 f32 WMMA (K=3 padded).
// One wave (32 lanes) per 16 candidates; loop over 64 tiles of 16 vote points.
__global__ __launch_bounds__(32) void d2seed_wmma_kernel(const float* __restrict__ wl4,
                                                         const float* __restrict__ vp,
                                                         int Mvp,
                                                         float* __restrict__ d2seed) {
    int lane = threadIdx.x;          // 0..31
    int half = lane >> 4;            // 0: K=0,1 ; 1: K=2,3 (A layout)
    int l    = lane & 15;
    int m0   = blockIdx.x * 16;

    // A tile (16 candidates x 4): lanes 0-15 -> (x,y), lanes 16-31 -> (z,0)
    int am = m0 + l;
    float ax = wl4[am * 4 + 0], ay = wl4[am * 4 + 1], az = wl4[am * 4 + 2];
    v2f a;
    a.x = half ? az : ax;
    a.y = half ? 0.0f : ay;

    float acc[8];
    #pragma unroll
    for (int v = 0; v < 8; ++v) acc[v] = 3.4e38f;

    int ntiles = Mvp >> 4;
    for (int tIdx = 0; tIdx < ntiles; ++tIdx) {
        int n = tIdx * 16 + l;       // this lane's column (both halves share N=l)
        float bx = vp[n * 3 + 0], by = vp[n * 3 + 1], bz = vp[n * 3 + 2];
        float vpn = bx * bx + by * by + bz * bz;     // ||b||^2 for column n
        v2f b;                                       // B tile (4 x 16)
        b.x = half ? bz : bx;
        b.y = half ? 0.0f : by;
        v8f c = {};
        c = __builtin_amdgcn_wmma_f32_16x16x4_f32(
                false, a, false, b, (short)0, c, false, false);
        #pragma unroll
        for (int v = 0; v < 8; ++v) {
            float val = vpn - 2.0f * c[v];           // ||b||^2 - 2 a.b
            acc[v] = fminf(acc[v], val);
        }
    }

    // min across the 16 lanes of each half (all have the same M set)
    #pragma unroll
    for (int off = 1; off < 16; off <<= 1) {
        #pragma unroll
        for (int v = 0; v < 8; ++v)
            acc[v] = fminf(acc[v], __shfl_xor(acc[v], off, 32));
    }

    if (l == 0) {                                    // lanes 0 and 16 hold results
        #pragma unroll
        for (int v = 0; v < 8; ++v) {
            int m = m0 + half * 8 + v;               // D layout: half0 M=0..7, half1 M=8..15
            float wx = wl4[m * 4 + 0], wy = wl4[m * 4 + 1], wz = wl4[m * 4 + 2];
            float d2 = acc[v] + wx * wx + wy * wy + wz * wz;
            d2seed[m] = sqrtf(fmaxf(d2, 0.0f));
        }
    }
}

// ---------------- 5) stable partition by keep, take first 256, write outputs ----------------
__global__ void select_kernel(const float* __restrict__ wl4,
                              const float* __restrict__ sc,
                              const float* __restrict__ d2seed,
                              float* __restrict__ out) {
    __shared__ int sel[NUM_PROP];
    if (threadIdx.x == 0) {
        int cnt = 0;
        for (int k = 0; k < KSAMP; ++k)
            if (d2seed[k] < SEED_THRESH) { if (cnt < NUM_PROP) sel[cnt] = k; cnt++; }
        if (cnt == 0) {                               // keep.any() fallback: keep all
            for (int k = 0; k < NUM_PROP; ++k) sel[k] = k;
        } else if (cnt < NUM_PROP) {                  // stable: keeps first, then rest
            int pos = cnt;
            for (int k = 0; k < KSAMP && pos < NUM_PROP; ++k)
                if (!(d2seed[k] < SEED_THRESH)) sel[pos++] = k;
        }
    }
    __syncthreads();
    int t = threadIdx.x;                              // 256 threads
    int k = sel[t];
    out[t * 3 + 0] = wl4[k * 4 + 0];                  // candidates [0, 768)
    out[t * 3 + 1] = wl4[k * 4 + 1];
    out[t * 3 + 2] = wl4[k * 4 + 2];
    out[NUM_PROP * 3 + t] = 0.0f;                     // probs [768, 1024)
    out[NUM_PROP * 4 + t * 3 + 0] = sc[k * 3 + 0];    // scales [1024, 1792)
    out[NUM_PROP * 4 + t * 3 + 1] = sc[k * 3 + 1];
    out[NUM_PROP * 4 + t * 3 + 2] = sc[k * 3 + 2];
}

// ---------------- host launcher ----------------
extern "C" void kernel_launch(void* const* d_in, const int* in_sizes, int n_in,
                              void* d_out, int out_size, void* d_ws, size_t ws_size,
                              hipStream_t stream) {
    const float* pc      = (const float*)d_in[0];
    const float* xyz     = (const float*)d_in[1];
    const float* scale   = (const float*)d_in[2];
    const float* prob    = (const float*)d_in[3];
    const float* corners = (const float*)d_in[4];
    const float* vp      = (const float*)d_in[5];
    float* out = (float*)d_out;

    int N   = in_sizes[0] / 3;       // 100000
    int Mvp = in_sizes[5] / 3;       // 1024

    const size_t NG = (size_t)GX * GY * GZ;   // 8,286,208 cells

    // workspace layout (~133 MB; accumulation surface is L2-resident)
    float* grid_obj = (float*)d_ws;
    float* gsx  = grid_obj + NG;
    float* gsy  = gsx + NG;
    float* gsz  = gsy + NG;
    float* dist = gsz + NG;                        // NXZ floats
    int*   yidx = (int*)(dist + NXZ);              // NXZ ints
    float* wl4  = (float*)(yidx + NXZ);            // KSAMP*4
    float* sc   = wl4 + (size_t)KSAMP * 4;         // KSAMP*3
    float* d2s  = sc + (size_t)KSAMP * 3;          // KSAMP

    // zero the vote accumulators (graph-capturable async memset)
    hipMemsetAsync(grid_obj, 0, NG * 4 * sizeof(float), stream);

    vote_kernel<<<(N + 255) / 256, 256, 0, stream>>>(
        pc, xyz, scale, prob, corners, grid_obj, gsx, gsy, gsz, N);

    reduce_y_kernel<<<(NXZ + 255) / 256, 256, 0, stream>>>(grid_obj, dist, yidx);

    sample_kernel<<<KSAMP, 256, 0, stream>>>(
        dist, yidx, corners, grid_obj, gsx, gsy, gsz, wl4, sc);

    d2seed_wmma_kernel<<<KSAMP / 16, 32, 0, stream>>>(wl4, vp, Mvp, d2s);

    select_kernel<<<1, NUM_PROP, 0, stream>>>(wl4, sc, d2s, out);
}